// FDCKernel_87462714016191
// MI455X (gfx1250) — compile-verified
//
#include <hip/hip_runtime.h>
#include <math.h>

typedef float v2f __attribute__((ext_vector_type(2)));
typedef float v8f __attribute__((ext_vector_type(8)));

#define B_ROWS   262144
#define OUT_DIM  256
#define COL_TILES 16               // 256 / 16
#define ROW_TILES_PER_WAVE 16
#define NUM_BLOCKS 2048            // 2048 blocks * 8 waves = 16384 waves
#define INV_2PI  0.15915493667125702f
#define INV_PI   0.3183098861837907f

// ---------------------------------------------------------------------------
// Prologue: zero the 512-float accumulator workspace (sum_real[256], sum_imag[256])
// ---------------------------------------------------------------------------
__global__ void fdc_zero_ws_kernel(float* __restrict__ ws) {
    int i = blockIdx.x * blockDim.x + threadIdx.x;
    if (i < 2 * OUT_DIM) ws[i] = 0.0f;
}

// ---------------------------------------------------------------------------
// Main: theta = x @ h.T + delta via V_WMMA_F32_16X16X4_F32 (K padded 2->4),
// then cos/sin, nontemporal store of cos, local accumulation of column sums.
// ---------------------------------------------------------------------------
__global__ __launch_bounds__(256) void fdc_main_kernel(
    const float* __restrict__ x, const float* __restrict__ h,
    const float* __restrict__ delta, float* __restrict__ out,
    float* __restrict__ ws)
{
    const int lane = threadIdx.x & 31;
    const int wave = threadIdx.x >> 5;
    const int gw   = blockIdx.x * 8 + wave;          // global wave id [0, 16384)

    const int col_tile = gw & (COL_TILES - 1);       // 16 column tiles
    const int col_base = col_tile * 16;
    const int row_tile_start = (gw >> 4) * ROW_TILES_PER_WAVE;

    const bool lo_half = (lane < 16);
    const int  nc  = lane & 15;
    const int  col = col_base + nc;

    // B operand (4x16, K x N), K=2 real + 2 zero-pad rows.
    // VGPR0: K=0 (lanes 0-15) / K=2 (lanes 16-31); VGPR1: K=1 / K=3.
    v2f bmat;
    if (lo_half) {
        bmat[0] = h[col * 2 + 0];
        bmat[1] = h[col * 2 + 1];
    } else {
        bmat[0] = 0.0f;
        bmat[1] = 0.0f;
    }
    const float dval = delta[col];

    // C/D layout: every VGPR of this lane maps to the SAME column `col`,
    // so preloading C with delta broadcasts the bias across the tile.
    v8f cinit;
#pragma unroll
    for (int v = 0; v < 8; ++v) cinit[v] = dval;

    float sum_r = 0.0f, sum_i = 0.0f;
    const float2* __restrict__ x2 = (const float2*)x;

    for (int rt = 0; rt < ROW_TILES_PER_WAVE; ++rt) {
        const int row_base = (row_tile_start + rt) * 16;

        // A operand (16x4, M x K): lanes 0-15 carry K=0/1, lanes 16-31 carry
        // the zero-padded K=2/3 plane.
        v2f amat;
        if (lo_half) {
            float2 xv = x2[row_base + lane];
            amat[0] = xv.x;
            amat[1] = xv.y;
        } else {
            amat[0] = 0.0f;
            amat[1] = 0.0f;
        }

        // theta_tile = A(16x4) * B(4x16) + delta  -> v_wmma_f32_16x16x4_f32
        v8f theta = __builtin_amdgcn_wmma_f32_16x16x4_f32(
            /*neg_a=*/false, amat, /*neg_b=*/false, bmat,
            /*c_mod=*/(short)0, cinit, /*reuse_a=*/false, /*reuse_b=*/false);

        // D layout: lanes 0-15 -> rows M=v, lanes 16-31 -> rows M=v+8.
        const int row0 = row_base + (lo_half ? 0 : 8);
        float* p = out + (size_t)row0 * OUT_DIM + col;
#pragma unroll
        for (int v = 0; v < 8; ++v) {
            const float t  = theta[v] * INV_2PI;       // HW sin/cos take turns
            const float cr = __builtin_amdgcn_cosf(t);
            const float si = __builtin_amdgcn_sinf(t);
            __builtin_nontemporal_store(cr, p);        // 268MB stream: bypass-ish
            p += OUT_DIM;
            sum_r += cr;
            sum_i += si;
        }
    }

    // Both half-waves accumulated the same 16 columns: fold, then one atomic
    // per column per wave.
    sum_r += __shfl_down(sum_r, 16, 32);
    sum_i += __shfl_down(sum_i, 16, 32);
    if (lo_half) {
        __hip_atomic_fetch_add(&ws[col], sum_r,
                               __ATOMIC_RELAXED, __HIP_MEMORY_SCOPE_AGENT);
        __hip_atomic_fetch_add(&ws[OUT_DIM + col], sum_i,
                               __ATOMIC_RELAXED, __HIP_MEMORY_SCOPE_AGENT);
    }
}

// ---------------------------------------------------------------------------
// Epilogue: eta = sum_c atan2(sum_i[c], sum_r[c]) / pi
// (atan2 is scale-invariant, so the 1/B mean normalization cancels)
// ---------------------------------------------------------------------------
__global__ void fdc_eta_kernel(const float* __restrict__ ws,
                               float* __restrict__ eta_out)
{
    __shared__ float red[OUT_DIM];
    const int c = threadIdx.x;
    red[c] = atan2f(ws[OUT_DIM + c], ws[c]);
    __syncthreads();
#pragma unroll
    for (int s = OUT_DIM / 2; s > 0; s >>= 1) {
        if (c < s) red[c] += red[c + s];
        __syncthreads();
    }
    if (c == 0) *eta_out = red[0] * INV_PI;
}

// ---------------------------------------------------------------------------
extern "C" void kernel_launch(void* const* d_in, const int* in_sizes, int n_in,
                              void* d_out, int out_size, void* d_ws, size_t ws_size,
                              hipStream_t stream) {
    const float* x     = (const float*)d_in[0];   // [262144, 2]
    const float* h     = (const float*)d_in[1];   // [256, 2]
    const float* delta = (const float*)d_in[2];   // [256]
    float* out = (float*)d_out;                   // [262144*256] ++ [eta]
    float* ws  = (float*)d_ws;                    // 512 floats of scratch
    float* eta = out + (out_size - 1);

    fdc_zero_ws_kernel<<<2, 256, 0, stream>>>(ws);
    fdc_main_kernel<<<NUM_BLOCKS, 256, 0, stream>>>(x, h, delta, out, ws);
    fdc_eta_kernel<<<1, OUT_DIM, 0, stream>>>(ws, eta);
}